// ControlUnit_11630771437684
// MI455X (gfx1250) — compile-verified
//
#include <hip/hip_runtime.h>

#define B_ 64
#define S_ 2048
#define D_ 512
#define SPLIT 8
#define ROWS_PER_SPLIT (S_ / SPLIT)   // 256

typedef __attribute__((ext_vector_type(2))) float v2f;
typedef __attribute__((ext_vector_type(4))) float f4;
typedef __attribute__((ext_vector_type(8))) float v8f;

#if __has_builtin(__builtin_amdgcn_wmma_f32_16x16x4_f32)
#define HAVE_WMMA_F32 1
#endif

// ---------------------------------------------------------------------------
// Phase 1: u[b,d] = (concat(c_i_1, q)[b,:] . W_d2d[d,:] + b_d[d]) * W_1d[d]
// GEMM M=64 (batch), N=512 (d), K=1024. One wave per 16x16 tile -> 128 waves.
// Uses V_WMMA_F32_16X16X4_F32 (f32 in, f32 accum), K stepped by 4.
// A 16x4 layout: lanes 0-15 -> M=lane, regs = K{0,1}; lanes 16-31 -> K{2,3}.
// B 4x16 layout mirrors A; C/D: reg r -> (M=r | M=r+8), N = lane&15.
// ---------------------------------------------------------------------------
__global__ __launch_bounds__(128) void phase1(const float* __restrict__ cprev,
                                              const float* __restrict__ q,
                                              const float* __restrict__ W,
                                              const float* __restrict__ bd,
                                              const float* __restrict__ w1,
                                              float* __restrict__ u) {
  const int lane  = threadIdx.x & 31;
  const int wglob = (blockIdx.x * blockDim.x + threadIdx.x) >> 5;  // 0..127
  const int mt = wglob >> 5;       // 0..3   batch tile
  const int nt = wglob & 31;       // 0..31  d tile
  const int m0 = mt * 16, n0 = nt * 16;
  const int kh  = lane >> 4;       // half-wave selects K pair
  const int l15 = lane & 15;
  const int row = m0 + l15;        // A row (batch index)
  const int col = n0 + l15;        // B col (d index)
  const float* __restrict__ wrow = W + (size_t)col * (2 * D_);  // W_d2d[col, :]

  v8f c = {0.f, 0.f, 0.f, 0.f, 0.f, 0.f, 0.f, 0.f};

#ifdef HAVE_WMMA_F32
  for (int k0 = 0; k0 < 2 * D_; k0 += 4) {
    const int ka = k0 + kh * 2;    // wave-uniform branch side (512 % 4 == 0)
    const float* asrc = (ka < D_) ? (cprev + row * D_ + ka)
                                  : (q     + row * D_ + (ka - D_));
    v2f a = *(const v2f*)asrc;            // 8B aligned (ka even)
    v2f b = *(const v2f*)(wrow + ka);
    c = __builtin_amdgcn_wmma_f32_16x16x4_f32(false, a, false, b,
                                              (short)0, c, false, false);
  }
#else
  // Scalar fallback (also what the host-side parse sees).
  for (int r = 0; r < 8; ++r) {
    const int mrow = m0 + r + kh * 8;
    float s = 0.f;
    for (int k = 0; k < D_; ++k) s = fmaf(cprev[mrow * D_ + k], wrow[k], s);
    for (int k = 0; k < D_; ++k) s = fmaf(q[mrow * D_ + k], wrow[D_ + k], s);
    c[r] = s;
  }
#endif

  const float scale = w1[col];
  const float bias  = bd[col];
  for (int r = 0; r < 8; ++r) {
    const int mrow = m0 + r + kh * 8;
    u[mrow * D_ + col] = (c[r] + bias) * scale;
  }
}

// ---------------------------------------------------------------------------
// Phase 2: streaming online softmax + weighted accumulation over a slice of S.
// One block per (split, batch); 8 waves; each wave handles one row at a time
// (lane L covers float4 chunks L, L+32, L+64, L+96 -> coalesced 512B loads).
// ---------------------------------------------------------------------------
__device__ __forceinline__ float dot4(f4 a, f4 b) {
  f4 p = a * b;
  return (p.x + p.y) + (p.z + p.w);
}

__global__ __launch_bounds__(256) void phase2(const float* __restrict__ cw,
                                              const float* __restrict__ u,
                                              float* __restrict__ pm,
                                              float* __restrict__ pl,
                                              float* __restrict__ pacc) {
  const int b    = blockIdx.y;
  const int sp   = blockIdx.x;
  const int tid  = threadIdx.x;
  const int wave = tid >> 5;
  const int lane = tid & 31;

  __shared__ float s_m[8];
  __shared__ float s_l[8];
  __shared__ float s_acc[D_];

  s_acc[tid]       = 0.f;
  s_acc[tid + 256] = 0.f;

  const f4* __restrict__ ub = (const f4*)(u + (size_t)b * D_);
  const f4 u0 = ub[lane], u1 = ub[lane + 32], u2 = ub[lane + 64], u3 = ub[lane + 96];

  const float* base = cw + ((size_t)b * S_ + (size_t)sp * ROWS_PER_SPLIT) * D_;

  float m = -3.402823466e38f;
  float l = 0.f;
  f4 a0 = {0.f, 0.f, 0.f, 0.f}, a1 = a0, a2 = a0, a3 = a0;

  for (int r = wave; r < ROWS_PER_SPLIT; r += 8) {
    const f4* row = (const f4*)(base + (size_t)r * D_);
    // Non-temporal: cw_s is 256MB streamed exactly once; keep it out of L2 RT set.
    f4 c0 = __builtin_nontemporal_load(row + lane);
    f4 c1 = __builtin_nontemporal_load(row + lane + 32);
    f4 c2 = __builtin_nontemporal_load(row + lane + 64);
    f4 c3 = __builtin_nontemporal_load(row + lane + 96);

    float p = dot4(c0, u0) + dot4(c1, u1) + dot4(c2, u2) + dot4(c3, u3);
    p += __shfl_xor(p, 16, 32);
    p += __shfl_xor(p, 8, 32);
    p += __shfl_xor(p, 4, 32);
    p += __shfl_xor(p, 2, 32);
    p += __shfl_xor(p, 1, 32);     // wave-uniform logit

    if (p > m) {                   // rare rescale path (wave-uniform branch)
      const float corr = __expf(m - p);
      l *= corr; a0 *= corr; a1 *= corr; a2 *= corr; a3 *= corr;
      m = p;
    }
    const float w = __expf(p - m);
    l += w;
    a0 += w * c0; a1 += w * c1; a2 += w * c2; a3 += w * c3;
  }

  if (lane == 0) { s_m[wave] = m; s_l[wave] = l; }
  __syncthreads();

  float mb = s_m[0];
  for (int j = 1; j < 8; ++j) mb = fmaxf(mb, s_m[j]);

  const float wscale = __expf(m - mb);   // uniform per wave
  f4 av[4] = {a0, a1, a2, a3};
  for (int i = 0; i < 4; ++i) {
    const int di = 4 * (lane + 32 * i);
    atomicAdd(&s_acc[di + 0], wscale * av[i].x);
    atomicAdd(&s_acc[di + 1], wscale * av[i].y);
    atomicAdd(&s_acc[di + 2], wscale * av[i].z);
    atomicAdd(&s_acc[di + 3], wscale * av[i].w);
  }
  __syncthreads();

  float lb = 0.f;
  for (int j = 0; j < 8; ++j) lb += s_l[j] * __expf(s_m[j] - mb);

  const int pidx = b * SPLIT + sp;
  if (tid == 0) { pm[pidx] = mb; pl[pidx] = lb; }
  pacc[(size_t)pidx * D_ + tid]       = s_acc[tid];
  pacc[(size_t)pidx * D_ + tid + 256] = s_acc[tid + 256];
}

// ---------------------------------------------------------------------------
// Phase 3: combine SPLIT partials per batch -> c_i[b, :]
// ---------------------------------------------------------------------------
__global__ __launch_bounds__(256) void phase3(const float* __restrict__ pm,
                                              const float* __restrict__ pl,
                                              const float* __restrict__ pacc,
                                              float* __restrict__ out) {
  const int b   = blockIdx.x;
  const int tid = threadIdx.x;

  float mb = -3.402823466e38f;
  for (int j = 0; j < SPLIT; ++j) mb = fmaxf(mb, pm[b * SPLIT + j]);

  float e[SPLIT];
  float lb = 0.f;
  for (int j = 0; j < SPLIT; ++j) {
    e[j] = __expf(pm[b * SPLIT + j] - mb);
    lb += pl[b * SPLIT + j] * e[j];
  }
  const float inv = 1.0f / lb;

  for (int d = tid; d < D_; d += 256) {
    float a = 0.f;
    for (int j = 0; j < SPLIT; ++j)
      a = fmaf(pacc[(size_t)(b * SPLIT + j) * D_ + d], e[j], a);
    out[b * D_ + d] = a * inv;
  }
}

// ---------------------------------------------------------------------------
extern "C" void kernel_launch(void* const* d_in, const int* in_sizes, int n_in,
                              void* d_out, int out_size, void* d_ws, size_t ws_size,
                              hipStream_t stream) {
  const float* c_i_1 = (const float*)d_in[0];
  const float* q     = (const float*)d_in[1];
  const float* cw_s  = (const float*)d_in[2];
  const float* W_d2d = (const float*)d_in[3];
  const float* b_d   = (const float*)d_in[4];
  const float* W_1d  = (const float*)d_in[5];
  // d_in[6] = b_1: uniform logit shift, cancels exactly in softmax.

  float* ws   = (float*)d_ws;
  float* u    = ws;                    // B*D            = 32768 floats
  float* pm   = u    + B_ * D_;        // B*SPLIT        = 512
  float* pl   = pm   + B_ * SPLIT;     // B*SPLIT        = 512
  float* pacc = pl   + B_ * SPLIT;     // B*SPLIT*D      = 262144  (~1.2 MB total)

  phase1<<<dim3(32), dim3(128), 0, stream>>>(c_i_1, q, W_d2d, b_d, W_1d, u);
  phase2<<<dim3(SPLIT, B_), dim3(256), 0, stream>>>(cw_s, u, pm, pl, pacc);
  phase3<<<dim3(B_), dim3(256), 0, stream>>>(pm, pl, pacc, (float*)d_out);
}